// DynamicChannelRouter_61495341744499
// MI455X (gfx1250) — compile-verified
//
#include <hip/hip_runtime.h>
#include <math.h>

// ---- problem constants (B,C,H,W fixed by the reference) ----
#define BB   16
#define CC   256
#define NN   4096          // H*W
#define CQ   64            // C/4
#define GG   8
#define CG   2048          // C*G
#define EPSF 1e-8f

typedef float v2f __attribute__((ext_vector_type(2)));
typedef float v8f __attribute__((ext_vector_type(8)));

// ---------------------------------------------------------------------------
// zero a small float buffer
__global__ void k_zero(float* p, int n) {
    int i = blockIdx.x * blockDim.x + threadIdx.x;
    if (i < n) p[i] = 0.0f;
}

// ---------------------------------------------------------------------------
// One block per (b,c) row of x: copy the 4096-float row to the output tail
// (the reference returns x verbatim) and compute max(||x_row||, eps).
__global__ void k_rownorm_copy(const float* __restrict__ x,
                               float* __restrict__ out_x,
                               float* __restrict__ xnorm) {
    __shared__ float red[8];
    int row = blockIdx.x;                       // b*256 + c, 4096 rows
    const float4* src = (const float4*)(x + (size_t)row * NN);
    float4*       dst = (float4*)(out_x + (size_t)row * NN);
    float ss = 0.0f;
    for (int i = threadIdx.x; i < NN / 4; i += 256) {
        float4 v = src[i];
        dst[i] = v;
        ss = fmaf(v.x, v.x, fmaf(v.y, v.y, fmaf(v.z, v.z, fmaf(v.w, v.w, ss))));
    }
    #pragma unroll
    for (int m = 16; m > 0; m >>= 1) ss += __shfl_xor(ss, m, 32);
    if ((threadIdx.x & 31) == 0) red[threadIdx.x >> 5] = ss;
    __syncthreads();
    if (threadIdx.x == 0) {
        float t = red[0];
        #pragma unroll
        for (int i = 1; i < 8; ++i) t += red[i];
        xnorm[row] = fmaxf(sqrtf(t), EPSF);
    }
}

// ---------------------------------------------------------------------------
// WMMA f32 16x16x4 conv GEMM:  h = relu(conv_w(64x256) @ x_b(256x4096) + b),
// immediately reduced over spatial columns into pooled-sum (atomic f32 add).
// Block: 256 threads = 8 waves; each wave owns one 16x16 D tile.
// Grid: (B, N/32); block covers M=64 (4 tiles) x N=32 (2 tiles).
__global__ void k_conv_pool(const float* __restrict__ x,
                            const float* __restrict__ convw,
                            const float* __restrict__ convb,
                            float* __restrict__ pooled) {
    __shared__ float lw[CQ * CC];               // 64 KB of the 320 KB WGP LDS
    for (int i = threadIdx.x; i < CQ * CC; i += 256) lw[i] = convw[i];
    __syncthreads();

    const int b    = blockIdx.x;
    const int wave = threadIdx.x >> 5;
    const int lane = threadIdx.x & 31;
    const int m0   = (wave & 3) * 16;                       // Cq tile
    const int n0   = blockIdx.y * 32 + (wave >> 2) * 16;    // spatial tile
    const int mn   = lane & 15;                             // row(A) / col(B)
    const int half = lane >> 4;                             // k-pair select

    v8f acc = {};
    const float* xb = x + ((size_t)b * CC) * NN + n0 + mn;  // B(k,n) base
    for (int k0 = 0; k0 < CC; k0 += 4) {
        // A 16x4 f32 frag: vgpr = k&1, lane-half = k>>1  ->  k = 2*half+{0,1}
        const float* wrow = &lw[(m0 + mn) * CC + k0 + 2 * half];
        v2f a; a.x = wrow[0]; a.y = wrow[1];
        // B 4x16 f32 frag: same (vgpr = k&1, half = k>>1) layout
        const float* bp = xb + (size_t)(k0 + 2 * half) * NN;
        v2f bf; bf.x = bp[0]; bf.y = bp[NN];
        acc = __builtin_amdgcn_wmma_f32_16x16x4_f32(
            false, a, false, bf, (short)0, acc, false, false);
    }

    // D: vgpr r -> rows m0+r (lanes 0-15) / m0+r+8 (lanes 16-31), col = lane&15.
    // bias + relu, then sum the 16 spatial columns and atomically fold into pooled.
    #pragma unroll
    for (int r = 0; r < 8; ++r) {
        int mrow = m0 + r + 8 * half;
        float v = fmaxf(acc[r] + convb[mrow], 0.0f);
        v += __shfl_xor(v, 1, 32);
        v += __shfl_xor(v, 2, 32);
        v += __shfl_xor(v, 4, 32);
        v += __shfl_xor(v, 8, 32);
        if (mn == 0) atomicAdd(&pooled[b * CQ + mrow], v);
    }
}

// ---------------------------------------------------------------------------
// logits[b,cg] = lin_b + lin_w @ (pooled/N)  +  log(ga + eps)
__global__ void k_logits(const float* __restrict__ pooled,
                         const float* __restrict__ linw,
                         const float* __restrict__ linb,
                         const float* __restrict__ ga,
                         float* __restrict__ logits) {
    int t  = blockIdx.x * blockDim.x + threadIdx.x;   // 32768
    int b  = t >> 11, cg = t & (CG - 1);
    const float inv = 1.0f / (float)NN;
    const float* wrow = linw + (size_t)cg * CQ;
    const float* prow = pooled + b * CQ;
    float acc = linb[cg];
    #pragma unroll 8
    for (int d = 0; d < CQ; ++d) acc = fmaf(wrow[d], prow[d] * inv, acc);
    logits[t] = acc + logf(ga[t] + EPSF);
}

// ---------------------------------------------------------------------------
// rowwise softmax over G=8; one thread per (b,c)
__global__ void k_softmax(const float* __restrict__ logits, float* __restrict__ out) {
    int row = blockIdx.x * blockDim.x + threadIdx.x;  // 4096
    const float* l = logits + row * GG;
    float m = l[0];
    #pragma unroll
    for (int g = 1; g < GG; ++g) m = fmaxf(m, l[g]);
    float e[GG], sum = 0.0f;
    #pragma unroll
    for (int g = 0; g < GG; ++g) { e[g] = expf(l[g] - m); sum += e[g]; }
    float inv = 1.0f / sum;
    #pragma unroll
    for (int g = 0; g < GG; ++g) out[row * GG + g] = e[g] * inv;
}

// ---------------------------------------------------------------------------
// s[b,g,n] = sum_c w[b,c,g] * x[b,c,n].  Block (b, n-chunk of 256); w row in LDS.
__global__ void k_group_s(const float* __restrict__ w,
                          const float* __restrict__ x,
                          float* __restrict__ s) {
    __shared__ float lwb[CG];                         // 8 KB
    int b = blockIdx.x;
    for (int i = threadIdx.x; i < CG; i += 256) lwb[i] = w[(size_t)b * CG + i];
    __syncthreads();
    int n = blockIdx.y * 256 + threadIdx.x;
    const float* xb = x + (size_t)b * CC * NN + n;
    float acc[GG] = {0, 0, 0, 0, 0, 0, 0, 0};
    for (int c = 0; c < CC; ++c) {
        float xv = xb[(size_t)c * NN];
        #pragma unroll
        for (int g = 0; g < GG; ++g) acc[g] = fmaf(lwb[c * GG + g], xv, acc[g]);
    }
    float* sb = s + (size_t)b * GG * NN + n;
    #pragma unroll
    for (int g = 0; g < GG; ++g) sb[(size_t)g * NN] = acc[g];
}

// ---------------------------------------------------------------------------
// s_norm[b,g] = max(||s[b,g,:]||, eps);  one block per (b,g)
__global__ void k_snorm(const float* __restrict__ s, float* __restrict__ snorm) {
    __shared__ float red[8];
    int row = blockIdx.x;                              // 128 rows
    const float* sr = s + (size_t)row * NN;
    float ss = 0.0f;
    for (int n = threadIdx.x; n < NN; n += 256) { float v = sr[n]; ss = fmaf(v, v, ss); }
    #pragma unroll
    for (int m = 16; m > 0; m >>= 1) ss += __shfl_xor(ss, m, 32);
    if ((threadIdx.x & 31) == 0) red[threadIdx.x >> 5] = ss;
    __syncthreads();
    if (threadIdx.x == 0) {
        float t = red[0];
        #pragma unroll
        for (int i = 1; i < 8; ++i) t += red[i];
        snorm[row] = fmaxf(sqrtf(t), EPSF);
    }
}

// ---------------------------------------------------------------------------
// logits[b,c,g] += (x[b,c,:].s[b,g,:]) / (xnorm*snorm).  One block per (b,c).
__global__ void k_sim_update(const float* __restrict__ x,
                             const float* __restrict__ s,
                             const float* __restrict__ xnorm,
                             const float* __restrict__ snorm,
                             float* __restrict__ logits) {
    __shared__ float red[8][GG];
    int row = blockIdx.x;                              // b*256 + c
    int b   = row >> 8;
    const float* xr = x + (size_t)row * NN;
    const float* sb = s + (size_t)b * GG * NN;
    float acc[GG] = {0, 0, 0, 0, 0, 0, 0, 0};
    for (int n = threadIdx.x; n < NN; n += 256) {
        float xv = xr[n];
        #pragma unroll
        for (int g = 0; g < GG; ++g) acc[g] = fmaf(xv, sb[(size_t)g * NN + n], acc[g]);
    }
    #pragma unroll
    for (int g = 0; g < GG; ++g) {
        #pragma unroll
        for (int m = 16; m > 0; m >>= 1) acc[g] += __shfl_xor(acc[g], m, 32);
    }
    int wave = threadIdx.x >> 5, lane = threadIdx.x & 31;
    if (lane == 0) {
        #pragma unroll
        for (int g = 0; g < GG; ++g) red[wave][g] = acc[g];
    }
    __syncthreads();
    if (threadIdx.x < GG) {
        int g = threadIdx.x;
        float dot = red[0][g];
        #pragma unroll
        for (int wv = 1; wv < 8; ++wv) dot += red[wv][g];
        logits[row * GG + g] += dot / (xnorm[row] * snorm[b * GG + g]);
    }
}

// ---------------------------------------------------------------------------
extern "C" void kernel_launch(void* const* d_in, const int* in_sizes, int n_in,
                              void* d_out, int out_size, void* d_ws, size_t ws_size,
                              hipStream_t stream) {
    const float* x     = (const float*)d_in[0];
    const float* ga    = (const float*)d_in[1];
    const float* convw = (const float*)d_in[2];
    const float* convb = (const float*)d_in[3];
    const float* linw  = (const float*)d_in[4];
    const float* linb  = (const float*)d_in[5];

    float* out_fw = (float*)d_out;                 // (B,C,G) = 32768
    float* out_x  = out_fw + BB * CC * GG;         // x copy, 16.7M floats

    // workspace layout (floats): ~2.4 MB total
    float* ws     = (float*)d_ws;
    float* xnorm  = ws;                            //  4096
    float* pooled = ws + 4096;                     //  1024
    float* logits = ws + 5120;                     // 32768
    float* wbuf   = ws + 37888;                    // 32768
    float* sbuf   = ws + 70656;                    // 524288  (B,G,N)
    float* snorm  = ws + 594944;                   //   128

    k_zero<<<4, 256, 0, stream>>>(pooled, BB * CQ);
    k_rownorm_copy<<<BB * CC, 256, 0, stream>>>(x, out_x, xnorm);
    k_conv_pool<<<dim3(BB, NN / 32), 256, 0, stream>>>(x, convw, convb, pooled);
    k_logits<<<(BB * CG) / 256, 256, 0, stream>>>(pooled, linw, linb, ga, logits);

    for (int it = 0; it < 2; ++it) {               // ROUTING_ITERS - 1 updates
        k_softmax<<<(BB * CC) / 256, 256, 0, stream>>>(logits, wbuf);
        k_group_s<<<dim3(BB, NN / 256), 256, 0, stream>>>(wbuf, x, sbuf);
        k_snorm<<<BB * GG, 256, 0, stream>>>(sbuf, snorm);
        k_sim_update<<<BB * CC, 256, 0, stream>>>(x, sbuf, xnorm, snorm, logits);
    }
    k_softmax<<<(BB * CC) / 256, 256, 0, stream>>>(logits, out_fw);
}